// Voxelization_80238579023887
// MI455X (gfx1250) — compile-verified
//
#include <hip/hip_runtime.h>

// Dynamic voxelization for MI455X (gfx1250).
// Memory-bound streaming kernel: 128 MB in, 96 MB out -> ~9.6 us floor @ 23.3 TB/s.
// Strategy: 4 points/thread, b128 coalesced loads/stores, non-temporal cache
// hints (one-touch stream, don't thrash the 192 MB L2), global_prefetch_b8.

typedef __attribute__((ext_vector_type(4))) float v4f;
typedef __attribute__((ext_vector_type(4))) int   v4i;

// Grid config (matches reference): voxel=(0.05,0.05,0.1), range x[0,70.4] y[-40,40] z[-3,1]
#define VSX 0.05f
#define VSY 0.05f
#define VSZ 0.10f
#define MINX 0.0f
#define MINY (-40.0f)
#define MINZ (-3.0f)
#define GX 1408
#define GY 1600
#define GZ 40

struct VoxCoord { int z, y, x; };

__device__ __forceinline__ VoxCoord voxel_one(float px, float py, float pz) {
    // Reference does float32 division then floor then int cast; keep exact fdiv.
    int ix = (int)__builtin_floorf((px - MINX) / VSX);
    int iy = (int)__builtin_floorf((py - MINY) / VSY);
    int iz = (int)__builtin_floorf((pz - MINZ) / VSZ);
    bool valid = (ix >= 0) & (ix < GX) &
                 (iy >= 0) & (iy < GY) &
                 (iz >= 0) & (iz < GZ);
    VoxCoord c;
    c.x = valid ? ix : -1;
    c.y = valid ? iy : -1;
    c.z = valid ? iz : -1;
    return c;
}

__global__ void __launch_bounds__(256)
voxelize_kernel(const v4f* __restrict__ pts4,   // one v4f per point: {x,y,z,feat}
                int* __restrict__ out,          // (3, N): rows z, y, x
                int n)                          // number of points
{
    const int q = blockIdx.x * blockDim.x + threadIdx.x;  // quad index
    const int base = q * 4;

    if (base + 3 < n) {
        // Prefetch a few KB ahead in the point stream (gfx1250 global_prefetch_b8).
        __builtin_prefetch((const void*)(pts4 + base + 1024), 0, 0);

        // Four coalesced b128 non-temporal loads (full rows; feat is free).
        v4f p0 = __builtin_nontemporal_load(pts4 + base + 0);
        v4f p1 = __builtin_nontemporal_load(pts4 + base + 1);
        v4f p2 = __builtin_nontemporal_load(pts4 + base + 2);
        v4f p3 = __builtin_nontemporal_load(pts4 + base + 3);

        VoxCoord c0 = voxel_one(p0.x, p0.y, p0.z);
        VoxCoord c1 = voxel_one(p1.x, p1.y, p1.z);
        VoxCoord c2 = voxel_one(p2.x, p2.y, p2.z);
        VoxCoord c3 = voxel_one(p3.x, p3.y, p3.z);

        v4i z4 = { c0.z, c1.z, c2.z, c3.z };
        v4i y4 = { c0.y, c1.y, c2.y, c3.y };
        v4i x4 = { c0.x, c1.x, c2.x, c3.x };

        // Three coalesced b128 non-temporal stores, one per output plane.
        __builtin_nontemporal_store(z4, (v4i*)(out + base));           // z row
        __builtin_nontemporal_store(y4, (v4i*)(out + n + base));       // y row
        __builtin_nontemporal_store(x4, (v4i*)(out + 2 * n + base));   // x row
    } else if (base < n) {
        // Scalar tail (not hit for N=8M, but keep the kernel general).
        for (int i = base; i < n; ++i) {
            v4f p = pts4[i];
            VoxCoord c = voxel_one(p.x, p.y, p.z);
            out[i]         = c.z;
            out[n + i]     = c.y;
            out[2 * n + i] = c.x;
        }
    }
}

extern "C" void kernel_launch(void* const* d_in, const int* in_sizes, int n_in,
                              void* d_out, int out_size, void* d_ws, size_t ws_size,
                              hipStream_t stream) {
    (void)n_in; (void)out_size; (void)d_ws; (void)ws_size;

    const v4f* pts4 = (const v4f*)d_in[0];     // [N,4] float32, 16B rows
    int* out = (int*)d_out;                     // (3, N) int32
    const int npts = in_sizes[0] / 4;

    const int quads = (npts + 3) / 4;
    const int block = 256;                      // 8 waves (wave32) per block
    const int grid = (quads + block - 1) / block;

    voxelize_kernel<<<grid, block, 0, stream>>>(pts4, out, npts);
}